// PolarVoxelization_40235253629493
// MI455X (gfx1250) — compile-verified
//
#include <hip/hip_runtime.h>
#include <math.h>

// -------------------------------------------------------------------------
// Polar voxelization, MI455X (gfx1250).
// Memory-bound streaming kernel: ~45 B HBM traffic per point, ~15.5 us floor
// at 23.3 TB/s. No matrix structure -> no WMMA. CDNA5-specific path used:
// async global->LDS staging (ASYNCcnt double buffering) per wave32.
// -------------------------------------------------------------------------

#define BLOCK   256
#define WAVE    32
#define WPB     (BLOCK / WAVE)
#define NRAD    64
#define NANG    512
#define Z_MIN_C (-3.0f)
#define Z_MAX_C (5.0f)

#if defined(__HIP_DEVICE_COMPILE__)
#  if __has_builtin(__builtin_amdgcn_global_load_async_to_lds_b128) && \
      __has_builtin(__builtin_amdgcn_s_wait_asynccnt)
#    define USE_ASYNC 1
#  else
#    define USE_ASYNC 0
#  endif
#else
#  define USE_ASYNC 0
#endif

#if USE_ASYNC
// Exact element type from the clang diagnostic:
//   'int __attribute__((__vector_size__(4 * sizeof(int)))) __device__ *'
typedef int v4i __attribute__((vector_size(4 * sizeof(int))));
typedef __attribute__((address_space(1))) v4i as1_v4i;   // printed as __device__
typedef __attribute__((address_space(3))) v4i as3_v4i;   // printed as __shared__

__device__ __forceinline__ void async_point_to_lds(const float* gsrc, float* lds_dst) {
  // 16 bytes per lane: one point (x,y,z,intensity) -> this lane's LDS slot.
  __builtin_amdgcn_global_load_async_to_lds_b128(
      (as1_v4i*)gsrc, (as3_v4i*)lds_dst, /*imm_offset=*/0, /*cpol=*/0);
}
#endif

// Exact searchsorted(edges, v, 'left') - 1, clipped: find j in [0, nbins-1]
// with edges[j] < v <= edges[j+1], starting from an analytic guess.
__device__ __forceinline__ void process_store(
    float4 p, long long idx,
    const float* __restrict__ s_red, const float* __restrict__ s_ang,
    float r_lo, float r_hi, float a_lo, float a_hi, float inv_da,
    float4* __restrict__ out_feat, int* __restrict__ out_coord,
    unsigned char* __restrict__ out_mask)
{
  const float x = p.x, y = p.y, z = p.z, w = p.w;
  const float r  = sqrtf(fmaf(x, x, y * y));
  const float th = atan2f(y, x);
  const bool m = (z >= Z_MIN_C) && (z < Z_MAX_C) &&
                 (r >= r_lo) && (r < r_hi) &&
                 (th >= a_lo) && (th < a_hi);

  // radial: edges[i] = 50*(i/64)^1.5  =>  i ~= 64*cbrt((r/50)^2)
  const float t = r * 0.02f;
  int j = (int)(64.0f * cbrtf(t * t));
  j = j < 0 ? 0 : (j > NRAD - 1 ? NRAD - 1 : j);
  while (j > 0 && r <= s_red[j]) --j;                 // need edges[j] < r
  while (j < NRAD - 1 && r > s_red[j + 1]) ++j;       // need edges[j+1] >= r

  // angular: uniform guess, then exact fix-up against float32 edges
  int k = (int)((th - a_lo) * inv_da);
  k = k < 0 ? 0 : (k > NANG - 1 ? NANG - 1 : k);
  while (k > 0 && th <= s_ang[k]) --k;
  while (k < NANG - 1 && th > s_ang[k + 1]) ++k;

  out_feat[idx] = m ? make_float4(x, y, z, w) : make_float4(0.f, 0.f, 0.f, 0.f);
  out_coord[idx * 3 + 0] = m ? j  : -1;
  out_coord[idx * 3 + 1] = m ? k  : -1;
  out_coord[idx * 3 + 2] = m ? 0  : -1;
  out_mask[idx] = m ? (unsigned char)1 : (unsigned char)0;
}

__global__ __launch_bounds__(BLOCK)
void PolarVoxelization_kernel(
    const float* __restrict__ points,
    const float* __restrict__ redges,
    const float* __restrict__ aedges,
    float4* __restrict__ out_feat,
    int* __restrict__ out_coord,
    unsigned char* __restrict__ out_mask,
    int n)
{
  __shared__ float s_red[NRAD + 1];
  __shared__ float s_ang[NANG + 1];
#if USE_ASYNC
  __shared__ __align__(16) float s_stage[WPB][2][WAVE * 4];
#endif

  for (int i = threadIdx.x; i < NRAD + 1; i += BLOCK) s_red[i] = redges[i];
  for (int i = threadIdx.x; i < NANG + 1; i += BLOCK) s_ang[i] = aedges[i];
  __syncthreads();

  const float r_lo = s_red[0];
  const float r_hi = s_red[NRAD];
  const float a_lo = s_ang[0];
  const float a_hi = s_ang[NANG];
  const float inv_da = (float)NANG / (a_hi - a_lo);

  const int lane = threadIdx.x & (WAVE - 1);
  const int wid  = threadIdx.x / WAVE;
  const long long gw   = (long long)blockIdx.x * WPB + wid;
  const long long nw   = (long long)gridDim.x * WPB;
  const long long step = nw * WAVE;
  const long long base = gw * WAVE;

#if USE_ASYNC
  float* stage[2] = { &s_stage[wid][0][0], &s_stage[wid][1][0] };
  if (base < n) {                                   // uniform per wave
    long long src = base + lane; if (src >= n) src = n - 1;
    async_point_to_lds(points + src * 4, stage[0] + lane * 4);
  }
  int buf = 0;
  for (long long i = base; i < n; i += step) {
    const long long inext = i + step;               // uniform per wave
    if (inext < n) {
      long long src = inext + lane; if (src >= n) src = n - 1;
      async_point_to_lds(points + src * 4, stage[buf ^ 1] + lane * 4);
      __builtin_amdgcn_s_wait_asynccnt(1);          // tile i complete, i+1 in flight
    } else {
      __builtin_amdgcn_s_wait_asynccnt(0);
    }
    asm volatile("" ::: "memory");                  // keep LDS reads after the wait
    const float4 p = *reinterpret_cast<const float4*>(stage[buf] + lane * 4);
    buf ^= 1;
    const long long idx = i + lane;
    if (idx < n) {
      process_store(p, idx, s_red, s_ang, r_lo, r_hi, a_lo, a_hi, inv_da,
                    out_feat, out_coord, out_mask);
    }
  }
#else
  for (long long i = base; i < n; i += step) {
    const long long idx = i + lane;
    if (idx < n) {
      const float4 p = reinterpret_cast<const float4*>(points)[idx];
      process_store(p, idx, s_red, s_ang, r_lo, r_hi, a_lo, a_hi, inv_da,
                    out_feat, out_coord, out_mask);
    }
  }
#endif
}

extern "C" void kernel_launch(void* const* d_in, const int* in_sizes, int n_in,
                              void* d_out, int out_size, void* d_ws, size_t ws_size,
                              hipStream_t stream) {
  const float* points = (const float*)d_in[0];   // (N,4) f32
  const float* redges = (const float*)d_in[1];   // (65,) f32
  const float* aedges = (const float*)d_in[2];   // (513,) f32
  const int n = in_sizes[0] / 4;

  // Outputs concatenated flat in return order: features f32 (N*4),
  // coords i32 (N*3), mask bool (N bytes).
  float4* out_feat = (float4*)d_out;
  int* out_coord = (int*)((float*)d_out + (size_t)n * 4);
  unsigned char* out_mask = (unsigned char*)(out_coord + (size_t)n * 3);

  int blocks = (n + BLOCK - 1) / BLOCK;
  if (blocks > 2048) blocks = 2048;   // grid-stride; keeps waves looping so the
  if (blocks < 1) blocks = 1;         // async double-buffer actually pipelines
  hipLaunchKernelGGL(PolarVoxelization_kernel, dim3(blocks), dim3(BLOCK), 0, stream,
                     points, redges, aedges, out_feat, out_coord, out_mask, n);
}